// GraphSAGECitationModel_1803886264537
// MI455X (gfx1250) — compile-verified
//
#include <hip/hip_runtime.h>
#include <math.h>

// ---------------------------------------------------------------------------
// GraphSAGE citation model for MI455X (gfx1250, wave32).
// Dense GEMMs use V_WMMA_F32_16X16X4_F32 (full f32 precision; GEMM FLOPs are
// negligible vs the L2-resident edge-scatter, so f32 matrix path is free).
// ---------------------------------------------------------------------------

typedef float v2f __attribute__((ext_vector_type(2)));
typedef float v8f __attribute__((ext_vector_type(8)));

#define N_NODES 50000
#define N_EDGES 1600000
#define IN_DIM  256
#define HID     128
#define BATCH   16384
#define BN_EPS  1e-5f

// ---------------------------------------------------------------------------
// Degree computation: one thread per edge, atomic count into deg[dst].
// ---------------------------------------------------------------------------
__global__ void deg_kernel(const int* __restrict__ dst, float* __restrict__ deg) {
    int t = blockIdx.x * blockDim.x + threadIdx.x;
    if (t < N_EDGES) atomicAdd(&deg[dst[t]], 1.0f);
}

// In-place deg -> 1/max(deg,1)
__global__ void rdeg_kernel(float* __restrict__ deg) {
    int t = blockIdx.x * blockDim.x + threadIdx.x;
    if (t < N_NODES) deg[t] = 1.0f / fmaxf(deg[t], 1.0f);
}

// ---------------------------------------------------------------------------
// Edge scatter: one wave per edge; lane l handles dims l, l+32, ...
// Gather of feat[src] rows is contiguous (128B per 32 lanes); the atomic adds
// land in L2 (agg arrays fit in the 192MB L2).
// ---------------------------------------------------------------------------
template<int DIM>
__global__ void scatter_kernel(const float* __restrict__ feat,
                               const int* __restrict__ src,
                               const int* __restrict__ dst,
                               float* __restrict__ agg) {
    int wave = (blockIdx.x * blockDim.x + threadIdx.x) >> 5;  // grid sized exactly
    int lane = threadIdx.x & 31;
    int s = src[wave];
    int d = dst[wave];
    const float* frow = feat + (size_t)s * DIM;
    float*       arow = agg  + (size_t)d * DIM;
#pragma unroll
    for (int i = 0; i < DIM; i += 32) {
        atomicAdd(&arow[i + lane], frow[i + lane]);
    }
}

// ---------------------------------------------------------------------------
// Fused SAGE layer GEMM:
//   out[m, :] = act( (agg[m,:] * rdeg[m]) @ wl + xin[m,:] @ wr + bias )
// One wave computes a 16x128 output tile with 8 f32 accumulators, looping the
// shared A fragment over 8 N-tiles per K-step (K=4 per WMMA).
// Launch: 5 waves/block, 625 blocks -> 3125 tiles = 50000 rows exactly.
// ---------------------------------------------------------------------------
template<int K, bool RELU>
__global__ void sage_gemm(const float* __restrict__ agg,
                          const float* __restrict__ rdeg,
                          const float* __restrict__ xin,
                          const float* __restrict__ wl,
                          const float* __restrict__ wr,
                          const float* __restrict__ bias,
                          float* __restrict__ out) {
    const int N = HID;
    int lane = threadIdx.x & 31;
    int tile = blockIdx.x * (blockDim.x >> 5) + (threadIdx.x >> 5);
    int m0 = tile * 16;
    int ml = lane & 15;      // row (A) / col (B)
    int kg = lane >> 4;      // K-pair group

    v8f acc[8] = {};

    float rd = rdeg[m0 + ml];
    const float* arow = agg + (size_t)(m0 + ml) * K;
    const float* xrow = xin + (size_t)(m0 + ml) * K;

    // mean-aggregated neighbors @ wl
    for (int k = 0; k < K; k += 4) {
        int kk = k + 2 * kg;
        v2f a = *(const v2f*)(arow + kk);
        a.x *= rd; a.y *= rd;
#pragma unroll
        for (int nt = 0; nt < 8; nt++) {
            int n = nt * 16 + ml;
            v2f b;
            b.x = wl[(size_t)kk * N + n];
            b.y = wl[(size_t)(kk + 1) * N + n];
            acc[nt] = __builtin_amdgcn_wmma_f32_16x16x4_f32(
                false, a, false, b, (short)0, acc[nt], false, false);
        }
    }
    // root features @ wr
    for (int k = 0; k < K; k += 4) {
        int kk = k + 2 * kg;
        v2f a = *(const v2f*)(xrow + kk);
#pragma unroll
        for (int nt = 0; nt < 8; nt++) {
            int n = nt * 16 + ml;
            v2f b;
            b.x = wr[(size_t)kk * N + n];
            b.y = wr[(size_t)(kk + 1) * N + n];
            acc[nt] = __builtin_amdgcn_wmma_f32_16x16x4_f32(
                false, a, false, b, (short)0, acc[nt], false, false);
        }
    }
    // epilogue: + bias, optional relu
#pragma unroll
    for (int nt = 0; nt < 8; nt++) {
        int col = nt * 16 + ml;
        float bb = bias[col];
#pragma unroll
        for (int v = 0; v < 8; v++) {
            int row = m0 + v + 8 * kg;
            float z = acc[nt][v] + bb;
            if (RELU) z = fmaxf(z, 0.0f);
            out[(size_t)row * N + col] = z;
        }
    }
}

// ---------------------------------------------------------------------------
// Fused link predictor: per 16-row tile
//   gather src/tgt embeddings, edge MLP (6->64, relu) into LDS,
//   z1 = relu([src|tgt|ep] @ p1_w + p1_b); BN(eval);   (WMMA, K=320)
//   z2 = relu(z1 @ p2_w + p2_b);                       (WMMA, K=128)
//   out = sigmoid(z2 @ p3_w + p3_b)                    (VALU dot, N=1)
// 4 waves / block, 256 blocks -> 1024 tiles = 16384 rows exactly.
// LDS per wave: zbuf 16x128 + ebuf 16x64 (z2 reuses ebuf) = 12KB.
// ---------------------------------------------------------------------------
__global__ void predictor_kernel(const float* __restrict__ h1,
                                 const int* __restrict__ srcn,
                                 const int* __restrict__ tgtn,
                                 const float* __restrict__ ef,
                                 const float* __restrict__ ep_w,
                                 const float* __restrict__ ep_b,
                                 const float* __restrict__ p1_w,
                                 const float* __restrict__ p1_b,
                                 const float* __restrict__ bn_g,
                                 const float* __restrict__ bn_b,
                                 const float* __restrict__ bn_m,
                                 const float* __restrict__ bn_v,
                                 const float* __restrict__ p2_w,
                                 const float* __restrict__ p2_b,
                                 const float* __restrict__ p3_w,
                                 const float* __restrict__ p3_b,
                                 float* __restrict__ out) {
    __shared__ float lds[4][16 * 128 + 16 * 64];

    int wv   = threadIdx.x >> 5;
    int lane = threadIdx.x & 31;
    int tile = blockIdx.x * 4 + wv;
    int m0 = tile * 16;
    int ml = lane & 15;
    int kg = lane >> 4;

    float* zbuf = &lds[wv][0];          // [16][128]
    float* ebuf = &lds[wv][16 * 128];   // [16][64], reused for z2

    // edge MLP: relu(ef @ ep_w + ep_b) -> ebuf[16][64]
    for (int idx = lane; idx < 16 * 64; idx += 32) {
        int m = idx >> 6;
        int j = idx & 63;
        float s = ep_b[j];
#pragma unroll
        for (int f = 0; f < 6; f++)
            s += ef[(size_t)(m0 + m) * 6 + f] * ep_w[f * 64 + j];
        ebuf[idx] = fmaxf(s, 0.0f);
    }
    __syncthreads();

    v8f acc[8] = {};
    const float* srow = h1 + (size_t)srcn[m0 + ml] * HID;
    const float* trow = h1 + (size_t)tgtn[m0 + ml] * HID;

    // K = [0,128): src embedding
    for (int k = 0; k < HID; k += 4) {
        int kk = k + 2 * kg;
        v2f a = *(const v2f*)(srow + kk);
#pragma unroll
        for (int nt = 0; nt < 8; nt++) {
            int n = nt * 16 + ml;
            v2f b;
            b.x = p1_w[(size_t)kk * HID + n];
            b.y = p1_w[(size_t)(kk + 1) * HID + n];
            acc[nt] = __builtin_amdgcn_wmma_f32_16x16x4_f32(
                false, a, false, b, (short)0, acc[nt], false, false);
        }
    }
    // K = [128,256): tgt embedding
    for (int k = 0; k < HID; k += 4) {
        int kk = k + 2 * kg;
        v2f a = *(const v2f*)(trow + kk);
#pragma unroll
        for (int nt = 0; nt < 8; nt++) {
            int n = nt * 16 + ml;
            v2f b;
            b.x = p1_w[(size_t)(128 + kk) * HID + n];
            b.y = p1_w[(size_t)(128 + kk + 1) * HID + n];
            acc[nt] = __builtin_amdgcn_wmma_f32_16x16x4_f32(
                false, a, false, b, (short)0, acc[nt], false, false);
        }
    }
    // K = [256,320): processed edge features (from LDS)
    for (int k = 0; k < 64; k += 4) {
        int kk = k + 2 * kg;
        v2f a;
        a.x = ebuf[ml * 64 + kk];
        a.y = ebuf[ml * 64 + kk + 1];
#pragma unroll
        for (int nt = 0; nt < 8; nt++) {
            int n = nt * 16 + ml;
            v2f b;
            b.x = p1_w[(size_t)(256 + kk) * HID + n];
            b.y = p1_w[(size_t)(256 + kk + 1) * HID + n];
            acc[nt] = __builtin_amdgcn_wmma_f32_16x16x4_f32(
                false, a, false, b, (short)0, acc[nt], false, false);
        }
    }
    // epilogue: relu then BatchNorm(eval) -> zbuf
#pragma unroll
    for (int nt = 0; nt < 8; nt++) {
        int col = nt * 16 + ml;
        float sc = bn_g[col] * rsqrtf(bn_v[col] + BN_EPS);
        float sh = bn_b[col] - bn_m[col] * sc;
        float bb = p1_b[col];
#pragma unroll
        for (int v = 0; v < 8; v++) {
            int row = v + 8 * kg;
            float z = fmaxf(acc[nt][v] + bb, 0.0f);
            zbuf[row * 128 + col] = z * sc + sh;
        }
    }
    __syncthreads();

    // z2 = relu(z1 @ p2_w + p2_b), K=128, N=64
    v8f acc2[4] = {};
    for (int k = 0; k < HID; k += 4) {
        int kk = k + 2 * kg;
        v2f a;
        a.x = zbuf[ml * 128 + kk];
        a.y = zbuf[ml * 128 + kk + 1];
#pragma unroll
        for (int nt = 0; nt < 4; nt++) {
            int n = nt * 16 + ml;
            v2f b;
            b.x = p2_w[(size_t)kk * 64 + n];
            b.y = p2_w[(size_t)(kk + 1) * 64 + n];
            acc2[nt] = __builtin_amdgcn_wmma_f32_16x16x4_f32(
                false, a, false, b, (short)0, acc2[nt], false, false);
        }
    }
#pragma unroll
    for (int nt = 0; nt < 4; nt++) {
        int col = nt * 16 + ml;
        float bb = p2_b[col];
#pragma unroll
        for (int v = 0; v < 8; v++)
            ebuf[(v + 8 * kg) * 64 + col] = fmaxf(acc2[nt][v] + bb, 0.0f);
    }
    __syncthreads();

    // final 64->1 dot + sigmoid (lanes 0..15, one row each)
    if (lane < 16) {
        float s = p3_b[0];
#pragma unroll
        for (int j = 0; j < 64; j++)
            s += ebuf[ml * 64 + j] * p3_w[j];
        out[m0 + ml] = 1.0f / (1.0f + __expf(-s));
    }
}

// ---------------------------------------------------------------------------
// Launch
// ---------------------------------------------------------------------------
extern "C" void kernel_launch(void* const* d_in, const int* in_sizes, int n_in,
                              void* d_out, int out_size, void* d_ws, size_t ws_size,
                              hipStream_t stream) {
    const float* x          = (const float*)d_in[0];
    const int*   edge_index = (const int*)d_in[1];
    const int*   e_src      = edge_index;
    const int*   e_dst      = edge_index + N_EDGES;
    const int*   src_nodes  = (const int*)d_in[2];
    const int*   tgt_nodes  = (const int*)d_in[3];
    const float* ef         = (const float*)d_in[4];
    const float* wl0        = (const float*)d_in[5];
    const float* b0         = (const float*)d_in[6];
    const float* wr0        = (const float*)d_in[7];
    const float* wl1        = (const float*)d_in[8];
    const float* b1         = (const float*)d_in[9];
    const float* wr1        = (const float*)d_in[10];
    const float* ep_w       = (const float*)d_in[11];
    const float* ep_b       = (const float*)d_in[12];
    const float* p1_w       = (const float*)d_in[13];
    const float* p1_b       = (const float*)d_in[14];
    const float* bn_g       = (const float*)d_in[15];
    const float* bn_b       = (const float*)d_in[16];
    const float* bn_m       = (const float*)d_in[17];
    const float* bn_v       = (const float*)d_in[18];
    const float* p2_w       = (const float*)d_in[19];
    const float* p2_b       = (const float*)d_in[20];
    const float* p3_w       = (const float*)d_in[21];
    const float* p3_b       = (const float*)d_in[22];
    float* out = (float*)d_out;

    // workspace layout (floats): agg0 | h0 | agg1 | h1 | rdeg  (~128.2 MB)
    float* agg0 = (float*)d_ws;
    float* h0   = agg0 + (size_t)N_NODES * IN_DIM;
    float* agg1 = h0   + (size_t)N_NODES * HID;
    float* h1   = agg1 + (size_t)N_NODES * HID;
    float* rdeg = h1   + (size_t)N_NODES * HID;

    // zero the accumulation buffers every launch (graph-replay safe)
    hipMemsetAsync(agg0, 0, (size_t)N_NODES * IN_DIM * sizeof(float), stream);
    hipMemsetAsync(agg1, 0, (size_t)N_NODES * HID * sizeof(float), stream);
    hipMemsetAsync(rdeg, 0, (size_t)N_NODES * sizeof(float), stream);

    deg_kernel<<<(N_EDGES + 255) / 256, 256, 0, stream>>>(e_dst, rdeg);
    rdeg_kernel<<<(N_NODES + 255) / 256, 256, 0, stream>>>(rdeg);

    // layer 0: scatter x into agg0, fused GEMM -> relu(h0)
    scatter_kernel<IN_DIM><<<N_EDGES / 8, 256, 0, stream>>>(x, e_src, e_dst, agg0);
    sage_gemm<IN_DIM, true><<<625, 160, 0, stream>>>(agg0, rdeg, x, wl0, wr0, b0, h0);

    // layer 1: scatter h0 into agg1, fused GEMM -> h1 (no relu)
    scatter_kernel<HID><<<N_EDGES / 8, 256, 0, stream>>>(h0, e_src, e_dst, agg1);
    sage_gemm<HID, false><<<625, 160, 0, stream>>>(agg1, rdeg, h0, wl1, wr1, b1, h1);

    // fused predictor
    predictor_kernel<<<BATCH / 16 / 4, 128, 0, stream>>>(
        h1, src_nodes, tgt_nodes, ef, ep_w, ep_b, p1_w, p1_b,
        bn_g, bn_b, bn_m, bn_v, p2_w, p2_b, p3_w, p3_b, out);
}